// Gardner_43061342109964
// MI455X (gfx1250) — compile-verified
//
#include <hip/hip_runtime.h>
#include <hip/hip_bf16.h>
#include <math.h>

// Problem constants (match reference)
#define NB      8
#define NN      131072
#define TLEN    (NN - 65 + 1)        // 131008 timing errors per row
#define OLEN    (TLEN - 32)          // 130976 shifted complex samples per row
#define NQ      (NN + 1)             // FIR / p_even sample count per row
#define NTILES  ((NQ + 255) / 256)   // 513 WMMA tiles of 256 outputs per row

typedef __attribute__((ext_vector_type(2))) float v2f;
typedef __attribute__((ext_vector_type(8))) float v8f;

__device__ __forceinline__ constexpr float kPI() { return 3.14159265358979323846f; }

// ---------------------------------------------------------------------------
// Kernel A: 32-tap FIR (odd taps of the half-band sinc) as Toeplitz GEMM on
// V_WMMA_F32_16X16X4_F32. One wave -> one 16x16 D tile -> 256 consecutive
// q[s] = |fir(x)[s]|^2 values. A[m,k] = x[base-16+16m+k] (K=48 accumulation),
// B[k,n] = g[k-n] (tap Toeplitz, built once per wave, reused for all tiles).
// ---------------------------------------------------------------------------
__global__ __launch_bounds__(256) void fir_q_kernel(const float* __restrict__ xr,
                                                    const float* __restrict__ xi,
                                                    float* __restrict__ q) {
    __shared__ float sRaw[34];          // [0..31] raw odd taps, [32] = 1/nrm
    __shared__ float sX[8][2][320];     // per-wave x window staging (re/im)

    const int tid  = threadIdx.x;
    const int wave = tid >> 5;
    const int lane = tid & 31;
    const int n    = lane & 15;         // column / row-in-tile index
    const int kh   = lane >> 4;         // K-half selector (ISA A/B layout)
    const int b    = blockIdx.y;

    // raw odd taps: h[2j+1] = sinc((2j+1-32)/2)/2 = sinc(j-15.5)/2
    if (tid < 32) {
        float z  = (float)tid - 15.5f;
        float pz = kPI() * z;
        sRaw[tid] = 0.5f * sinf(pz) / pz;
    }
    __syncthreads();
    if (tid == 0) {
        float ss = 0.25f;               // center tap (1/2)^2
        for (int j = 0; j < 32; ++j) ss += sRaw[j] * sRaw[j];
        sRaw[32] = 1.0f / sqrtf(ss);    // normalization scale
    }
    __syncthreads();
    const float scale = sRaw[32];

    // B (Toeplitz of taps) is constant: build its 12 K-chunks once per wave.
    // g[k] = hh[31-k] so that out[s] = sum_k g[k]*x[s-16+k].
    v2f Bc[12];
#pragma unroll
    for (int kc = 0; kc < 12; ++kc) {
        int i0 = 4 * kc + 2 * kh - n;
        int i1 = i0 + 1;
        float b0 = (i0 >= 0 && i0 < 32) ? sRaw[31 - i0] * scale : 0.0f;
        float b1 = (i1 >= 0 && i1 < 32) ? sRaw[31 - i1] * scale : 0.0f;
        Bc[kc] = (v2f){b0, b1};
    }

    const int  tile  = blockIdx.x * 8 + wave;
    const bool valid = (tile < NTILES);
    const int  base  = tile * 256;
    const float* xrb = xr + (size_t)b * NN;
    const float* xib = xi + (size_t)b * NN;

    // Stage the 304-float x window (x[base-16 .. base+287]) per wave.
    if (valid) {
        for (int i = lane; i < 304; i += 32) {
            int gi = base - 16 + i;
            bool ok = (gi >= 0) && (gi < NN);
            sX[wave][0][i] = ok ? xrb[gi] : 0.0f;
            sX[wave][1][i] = ok ? xib[gi] : 0.0f;
        }
    }
    __syncthreads();

    if (valid) {
        v8f Dr = {0.f,0.f,0.f,0.f,0.f,0.f,0.f,0.f};
        v8f Di = {0.f,0.f,0.f,0.f,0.f,0.f,0.f,0.f};
        const float* lr = &sX[wave][0][0];
        const float* li = &sX[wave][1][0];
#pragma unroll
        for (int kc = 0; kc < 12; ++kc) {
            int o = 16 * n + 4 * kc + 2 * kh;          // A: M=lane%16, K=4kc+2kh+v
            v2f Ar = (v2f){lr[o], lr[o + 1]};
            v2f Ai = (v2f){li[o], li[o + 1]};
            Dr = __builtin_amdgcn_wmma_f32_16x16x4_f32(false, Ar, false, Bc[kc],
                                                       (short)0, Dr, false, false);
            Di = __builtin_amdgcn_wmma_f32_16x16x4_f32(false, Ai, false, Bc[kc],
                                                       (short)0, Di, false, false);
        }
        // D layout: VGPR d holds row M = d + 8*kh, column = lane%16.
        float* qb = q + (size_t)b * NQ;
#pragma unroll
        for (int d = 0; d < 8; ++d) {
            int s = base + 16 * (d + 8 * kh) + n;
            if (s < NQ) qb[s] = Dr[d] * Dr[d] + Di[d] * Di[d];
        }
    }
}

// ---------------------------------------------------------------------------
// Kernel B: err[s] = (q[s]-q[s+1]) * c^2*|x[s]|^2, then 65-wide moving sum
// (/32) -> timing_errors. 256 t's per block, 320 err values staged in LDS.
// ---------------------------------------------------------------------------
__global__ __launch_bounds__(256) void te_kernel(const float* __restrict__ xr,
                                                 const float* __restrict__ xi,
                                                 const float* __restrict__ q,
                                                 float* __restrict__ te_out) {
    __shared__ float sErr[320];
    __shared__ float sC2;

    const int tid = threadIdx.x;
    const int b   = blockIdx.y;

    if (tid == 0) {
        float ss = 0.25f;
        for (int j = 0; j < 32; ++j) {
            float z  = (float)j - 15.5f;
            float pz = kPI() * z;
            float r  = 0.5f * sinf(pz) / pz;
            ss += r * r;
        }
        sC2 = 0.25f / ss;               // (center_tap/nrm)^2
    }
    __syncthreads();
    const float c2 = sC2;

    const float* xrb = xr + (size_t)b * NN;
    const float* xib = xi + (size_t)b * NN;
    const float* qb  = q  + (size_t)b * NQ;
    const int t0 = blockIdx.x * 256;

    for (int i = tid; i < 320; i += 256) {
        int s = t0 + i;
        float e = 0.0f;
        if (s < NN) {
            float px = xrb[s], py = xib[s];
            e = (qb[s] - qb[s + 1]) * c2 * (px * px + py * py);
        }
        sErr[i] = e;
    }
    __syncthreads();

    const int t = t0 + tid;
    if (t < TLEN) {
        float acc = 0.0f;
        for (int u = 0; u < 65; ++u) acc += sErr[tid + u];
        te_out[(size_t)b * TLEN + t] = acc * (1.0f / 32.0f);
    }
}

// ---------------------------------------------------------------------------
// Kernel C: out[t'] = sum_k x[t'+32+k] * sinc(k-16-te[t'+16]).
// sinc(a-e) = -(-1)^a * sin(pi*e)/(pi*(a-e)) for integer a -> 1 sinf + 33 divs.
// ---------------------------------------------------------------------------
__global__ __launch_bounds__(256) void interp_kernel(const float* __restrict__ xr,
                                                     const float* __restrict__ xi,
                                                     const float* __restrict__ te,
                                                     float* __restrict__ outc) {
    __shared__ float sXr[288];
    __shared__ float sXi[288];

    const int tid = threadIdx.x;
    const int b   = blockIdx.y;
    const int t0  = blockIdx.x * 256;
    const float* xrb = xr + (size_t)b * NN;
    const float* xib = xi + (size_t)b * NN;

    for (int i = tid; i < 288; i += 256) {
        int gi = t0 + 32 + i;
        bool ok = gi < NN;
        sXr[i] = ok ? xrb[gi] : 0.0f;
        sXi[i] = ok ? xib[gi] : 0.0f;
    }
    __syncthreads();

    const int t = t0 + tid;
    if (t < OLEN) {
        float e  = te[(size_t)b * TLEN + t + 16];
        float sp = sinf(kPI() * e);
        float sr = 0.0f, si = 0.0f;
        float sgn = -1.0f;              // k=0 -> a=-16 (even) -> -(-1)^a = -1
#pragma unroll
        for (int k = 0; k < 33; ++k) {
            float d = (float)(k - 16) - e;
            float f = (fabsf(d) < 1e-5f) ? 1.0f : sgn * sp / (kPI() * d);
            sr += sXr[tid + k] * f;
            si += sXi[tid + k] * f;
            sgn = -sgn;
        }
        size_t o = ((size_t)b * OLEN + t) * 2;   // complex64: interleaved re,im
        outc[o]     = sr;
        outc[o + 1] = si;
    }
}

extern "C" void kernel_launch(void* const* d_in, const int* in_sizes, int n_in,
                              void* d_out, int out_size, void* d_ws, size_t ws_size,
                              hipStream_t stream) {
    const float* xr = (const float*)d_in[0];   // inputs_real  [8][131072]
    const float* xi = (const float*)d_in[1];   // inputs_imag  [8][131072]
    float* out = (float*)d_out;
    // d_out layout: shifted (8 x 130976 complex64, interleaved) then
    //               timing_errors (8 x 131008 f32)
    float* te_out = out + (size_t)NB * OLEN * 2;
    float* q      = (float*)d_ws;              // NB * (NN+1) f32 scratch

    dim3 blk(256, 1, 1);
    fir_q_kernel<<<dim3((NTILES + 7) / 8, NB), blk, 0, stream>>>(xr, xi, q);
    te_kernel<<<dim3((TLEN + 255) / 256, NB), blk, 0, stream>>>(xr, xi, q, te_out);
    interp_kernel<<<dim3((OLEN + 255) / 256, NB), blk, 0, stream>>>(xr, xi, te_out, out);
}